// DeeperGraphNet_12189117186686
// MI455X (gfx1250) — compile-verified
//
#include <hip/hip_runtime.h>
#include <hip/hip_bf16.h>

// Problem constants (match reference)
#define B_    32
#define N_    4096
#define BN_   (B_ * N_)
#define DIMV  64
#define DIN0  16

typedef float v2f __attribute__((ext_vector_type(2)));
typedef float v8f __attribute__((ext_vector_type(8)));

typedef unsigned int u32x4 __attribute__((ext_vector_type(4)));
typedef int          i32x4 __attribute__((ext_vector_type(4)));
typedef int          i32x8 __attribute__((ext_vector_type(8)));

#if defined(__gfx1250__) && __has_builtin(__builtin_amdgcn_tensor_load_to_lds)
#define USE_TDM 1
#else
#define USE_TDM 0
#endif

#if USE_TDM
// ---------------------------------------------------------------------------
// Tensor Data Mover: DMA a 2D row-major fp32 tile [rows x cols] (row stride =
// cols elements) from global memory into LDS.  D# built per CDNA5 ISA ch.8:
//   group0: count=1 | lds_addr | global_addr | type=2
//   group1: data_size=4B | tensor_dim0/1 | tile_dim0/1 | tensor_dim0_stride
// Issued by one wave; completion via s_wait_tensorcnt (TENSORcnt).
// ---------------------------------------------------------------------------
__device__ inline void tdm_load_2d(const float* gptr, float* lptr, int rows, int cols) {
  unsigned long long ga = (unsigned long long)gptr;
  unsigned la = (unsigned)(unsigned long long)lptr;  // low 32 bits = LDS offset
  u32x4 g0;
  g0[0] = 1u;                                            // count=1, user mode
  g0[1] = la;                                            // lds_addr
  g0[2] = (unsigned)(ga & 0xFFFFFFFFull);                // global_addr[31:0]
  g0[3] = (unsigned)((ga >> 32) & 0x01FFFFFFull) | (2u << 30);  // addr[56:32] | type=2
  unsigned ucols = (unsigned)cols, urows = (unsigned)rows;
  i32x8 g1;
  g1[0] = (int)(2u << 16);                               // workgroup_mask=0, data_size=4B
  g1[1] = (int)((ucols & 0xFFFFu) << 16);                // tensor_dim0[15:0]
  g1[2] = (int)((ucols >> 16) | ((urows & 0xFFFFu) << 16));   // dim0[31:16] | dim1[15:0]
  g1[3] = (int)((urows >> 16) | ((ucols & 0xFFFFu) << 16));   // dim1[31:16] | tile_dim0
  g1[4] = (int)(urows & 0xFFFFu);                        // tile_dim1 | tile_dim2=0
  g1[5] = cols;                                          // tensor_dim0_stride[31:0]
  g1[6] = 0;                                             // stride0 hi | dim1_stride lo
  g1[7] = 0;
  i32x4 z4 = {0, 0, 0, 0};
#if __clang_major__ >= 23
  i32x8 z8 = {0, 0, 0, 0, 0, 0, 0, 0};
  __builtin_amdgcn_tensor_load_to_lds(g0, g1, z4, z4, z8, 0);
#else
  __builtin_amdgcn_tensor_load_to_lds(g0, g1, z4, z4, 0);
#endif
}
#endif

// ---------------------------------------------------------------------------
// Init: nmask = 1 everywhere, acc = 0
// ---------------------------------------------------------------------------
__global__ void init_state_kernel(int* __restrict__ nmask, float* __restrict__ acc) {
  int i = blockIdx.x * blockDim.x + threadIdx.x;
  if (i < BN_) nmask[i] = 1;
  if (i < B_ * 2 * DIMV) acc[i] = 0.0f;
}

__global__ void zero_aggdeg_kernel(float* __restrict__ agg, float* __restrict__ deg) {
  int i = blockIdx.x * blockDim.x + threadIdx.x;
  if (i < BN_ * DIMV) agg[i] = 0.0f;
  if (i < BN_) deg[i] = 0.0f;
}

// ---------------------------------------------------------------------------
// SAGE mean-aggregation: edge-parallel, float4 gathers + fp32 global atomics.
// emask == nmask[src] && nmask[dst] (masks only shrink; see analysis)
// ---------------------------------------------------------------------------
template <int DIN>
__global__ void sage_aggregate_kernel(const int* __restrict__ src, const int* __restrict__ dst,
                                      const float* __restrict__ h, const int* __restrict__ nmask,
                                      float* __restrict__ agg, float* __restrict__ deg,
                                      int nedges) {
  int e = blockIdx.x * blockDim.x + threadIdx.x;
  if (e >= nedges) return;
  int s = src[e];
  int d = dst[e];
  if (!nmask[s] || !nmask[d]) return;
  atomicAdd(&deg[d], 1.0f);
  const float4* hs4 = (const float4*)(h + (long)s * DIN);
  float* ad = agg + (long)d * DIN;
#pragma unroll
  for (int j = 0; j < DIN / 4; ++j) {
    float4 v = hs4[j];
    atomicAdd(&ad[4 * j + 0], v.x);
    atomicAdd(&ad[4 * j + 1], v.y);
    atomicAdd(&ad[4 * j + 2], v.z);
    atomicAdd(&ad[4 * j + 3], v.w);
  }
}

// ---------------------------------------------------------------------------
// Fused SAGE linear: hout = relu( (agg/max(deg,1)) @ Wl + bl + hin @ Wr )
// One wave (32 lanes) computes a 16xDIMV output tile via v_wmma_f32_16x16x4_f32
// chained along K. Weights staged in LDS via the Tensor Data Mover.
//
// fp32 WMMA fragment layout (16x16x4):
//   A (16x4):  lane l: m = l%16, h = l/16 ; vgpr0 = A[m][k0+2h], vgpr1 = A[m][k0+2h+1]
//   B (4x16):  lane l: n = l%16, h = l/16 ; vgpr0 = B[k0+2h][n], vgpr1 = B[k0+2h+1][n]
//   C/D (16x16): vgpr r: lanes 0-15 -> (M=r, N=l), lanes 16-31 -> (M=r+8, N=l-16)
// ---------------------------------------------------------------------------
template <int DIN>
__global__ __launch_bounds__(256)
void sage_linear_kernel(const float* __restrict__ hin, const float* __restrict__ agg,
                        const float* __restrict__ deg,
                        const float* __restrict__ Wl, const float* __restrict__ bl,
                        const float* __restrict__ Wr,
                        float* __restrict__ hout) {
  __shared__ float lWl[DIMV * DIMV];
  __shared__ float lWr[DIMV * DIMV];
  __shared__ float lb[DIMV];

  int tid = threadIdx.x;
#if USE_TDM
  if (tid < 32) {  // wave 0 issues both tensor DMAs and waits on TENSORcnt
    tdm_load_2d(Wl, lWl, DIN, DIMV);
    tdm_load_2d(Wr, lWr, DIN, DIMV);
    __builtin_amdgcn_s_wait_tensorcnt(0);
  }
#else
  for (int i = tid; i < DIN * DIMV; i += 256) {
    lWl[i] = Wl[i];
    lWr[i] = Wr[i];
  }
#endif
  if (tid < DIMV) lb[tid] = bl[tid];
  __syncthreads();

  int wave = tid >> 5;
  int lane = tid & 31;
  int m  = lane & 15;     // A row index
  int hh = lane >> 4;     // which K-half of the fragment
  int n  = m;             // B/C column for this lane

  long rowbase = (long)blockIdx.x * 128 + (long)wave * 16;
  long node = rowbase + m;

  float inv = 1.0f / fmaxf(deg[node], 1.0f);
  const float* arow = agg + node * DIN;
  const float* xrow = hin + node * DIN;
  __builtin_prefetch(arow, 0, 1);
  __builtin_prefetch(xrow, 0, 1);

#pragma unroll
  for (int t = 0; t < 4; ++t) {          // 4 tiles of 16 output columns
    float bsv = lb[t * 16 + n];
    v8f c;
#pragma unroll
    for (int i = 0; i < 8; ++i) c[i] = bsv;

    // (agg/deg) @ Wl
#pragma unroll
    for (int k0 = 0; k0 < DIN; k0 += 4) {
      v2f a, b;
      a.x = arow[k0 + 2 * hh] * inv;
      a.y = arow[k0 + 2 * hh + 1] * inv;
      b.x = lWl[(k0 + 2 * hh) * DIMV + t * 16 + n];
      b.y = lWl[(k0 + 2 * hh + 1) * DIMV + t * 16 + n];
      c = __builtin_amdgcn_wmma_f32_16x16x4_f32(false, a, false, b, (short)0, c, false, false);
    }
    // x @ Wr
#pragma unroll
    for (int k0 = 0; k0 < DIN; k0 += 4) {
      v2f a, b;
      a.x = xrow[k0 + 2 * hh];
      a.y = xrow[k0 + 2 * hh + 1];
      b.x = lWr[(k0 + 2 * hh) * DIMV + t * 16 + n];
      b.y = lWr[(k0 + 2 * hh + 1) * DIMV + t * 16 + n];
      c = __builtin_amdgcn_wmma_f32_16x16x4_f32(false, a, false, b, (short)0, c, false, false);
    }

    // ReLU + store (lanes 0-15 write row i, lanes 16-31 write row i+8)
#pragma unroll
    for (int i = 0; i < 8; ++i) {
      float v = c[i];
      v = v > 0.0f ? v : 0.0f;
      hout[(rowbase + i + 8 * hh) * DIMV + t * 16 + n] = v;
    }
  }
}

// ---------------------------------------------------------------------------
// TopK scoring: s = tanh(x.p/|p|); key = order-preserving uint (0 for dead)
// ---------------------------------------------------------------------------
__global__ void topk_score_kernel(const float* __restrict__ h, const float* __restrict__ pw,
                                  const int* __restrict__ nmask,
                                  float* __restrict__ s, unsigned* __restrict__ key) {
  int i = blockIdx.x * blockDim.x + threadIdx.x;
  if (i >= BN_) return;
  const float4* p4 = (const float4*)pw;
  const float4* r4 = (const float4*)(h + (long)i * DIMV);
  float nrm = 0.0f, dot = 0.0f;
#pragma unroll
  for (int j = 0; j < DIMV / 4; ++j) {
    float4 p = p4[j];
    float4 r = r4[j];
    nrm += p.x * p.x + p.y * p.y + p.z * p.z + p.w * p.w;
    dot += r.x * p.x + r.y * p.y + r.z * p.z + r.w * p.w;
  }
  float sc = tanhf(dot / sqrtf(nrm));
  s[i] = sc;
  unsigned u = __float_as_uint(sc);
  u = (u & 0x80000000u) ? ~u : (u | 0x80000000u);  // monotonic float->uint
  key[i] = nmask[i] ? u : 0u;                       // dead nodes sink to minimum
}

// ---------------------------------------------------------------------------
// Per-graph 4-pass MSB radix select: find k-th largest key T and how many
// equal-to-T entries to accept (kEq). One block per graph.
// ---------------------------------------------------------------------------
__global__ __launch_bounds__(256)
void topk_radix_kernel(const unsigned* __restrict__ key, int kkeep,
                       unsigned* __restrict__ thresh, int* __restrict__ keqArr) {
  __shared__ unsigned hist[256];
  __shared__ unsigned sPref;
  __shared__ int sRem;
  int g = blockIdx.x, tid = threadIdx.x;
  if (tid == 0) { sPref = 0u; sRem = kkeep; }
  __syncthreads();
  for (int pass = 3; pass >= 0; --pass) {
    hist[tid] = 0u;
    __syncthreads();
    unsigned pm = (pass == 3) ? 0u : (~0u << ((pass + 1) * 8));
    unsigned pref = sPref;
    for (int i = tid; i < N_; i += 256) {
      unsigned u = key[g * N_ + i];
      if ((u & pm) == (pref & pm)) atomicAdd(&hist[(u >> (pass * 8)) & 255u], 1u);
    }
    __syncthreads();
    if (tid == 0) {
      int cum = 0, rem = sRem, chosen = 0;
      for (int b = 255; b >= 0; --b) {
        int c = (int)hist[b];
        if (cum + c >= rem) { chosen = b; rem -= cum; break; }
        cum += c;
      }
      sPref |= ((unsigned)chosen) << (pass * 8);
      sRem = rem;
    }
    __syncthreads();
  }
  if (tid == 0) { thresh[g] = sPref; keqArr[g] = sRem; }
}

// ---------------------------------------------------------------------------
// Stable selection + apply: keep keys > T, plus first kEq keys == T in index
// order (deterministic block prefix-scan). x_kept *= s, dropped -> 0.
// ---------------------------------------------------------------------------
__global__ __launch_bounds__(256)
void topk_apply_kernel(const unsigned* __restrict__ key,
                       const unsigned* __restrict__ thresh, const int* __restrict__ keqArr,
                       const float* __restrict__ s,
                       float* __restrict__ h, int* __restrict__ nmask) {
  __shared__ int sc[256];
  __shared__ int takenEq;
  int g = blockIdx.x, tid = threadIdx.x;
  unsigned T = thresh[g];
  int kEq = keqArr[g];
  if (tid == 0) takenEq = 0;
  __syncthreads();
  for (int base = 0; base < N_; base += 256) {
    int idx = g * N_ + base + tid;
    unsigned u = key[idx];
    int eq = (u == T) ? 1 : 0;
    int greater = (u > T) ? 1 : 0;
    sc[tid] = eq;
    __syncthreads();
    for (int off = 1; off < 256; off <<= 1) {
      int v = (tid >= off) ? sc[tid - off] : 0;
      __syncthreads();
      sc[tid] += v;
      __syncthreads();
    }
    int excl = sc[tid] - eq;
    int tot = sc[255];
    int baseTaken = takenEq;
    int sel = greater || (eq && (baseTaken + excl) < kEq);
    nmask[idx] = sel;
    float sv = s[idx];
    float* row = h + (long)idx * DIMV;
    if (sel) {
      for (int j = 0; j < DIMV; ++j) row[j] *= sv;
    } else {
      for (int j = 0; j < DIMV; ++j) row[j] = 0.0f;
    }
    __syncthreads();
    if (tid == 0) takenEq += tot;
    __syncthreads();
  }
}

// ---------------------------------------------------------------------------
// Readout: acc[g] += [masked max || sum/kkeep]; 64 threads = 64 dims per graph
// ---------------------------------------------------------------------------
__global__ void readout_kernel(const float* __restrict__ h, const int* __restrict__ nmask,
                               float* __restrict__ acc, float inv_k) {
  int g = blockIdx.x, d = threadIdx.x;
  float mx = -1e30f, sm = 0.0f;
  for (int i = 0; i < N_; ++i) {
    long idx = (long)g * N_ + i;
    float v = h[idx * DIMV + d];
    if (nmask[idx]) mx = fmaxf(mx, v);
    sm += v;  // dropped rows already zeroed
  }
  acc[g * 2 * DIMV + d] += mx;
  acc[g * 2 * DIMV + DIMV + d] += sm * inv_k;
}

// ---------------------------------------------------------------------------
// Final MLP: out = relu(acc @ W1 + b1) @ W2 + b2   ([32,128]->[32,64]->[32,6])
// ---------------------------------------------------------------------------
__global__ __launch_bounds__(256)
void mlp_kernel(const float* __restrict__ acc,
                const float* __restrict__ W1, const float* __restrict__ b1,
                const float* __restrict__ W2, const float* __restrict__ b2,
                float* __restrict__ out) {
  __shared__ float hid[B_ * DIMV];
  int tid = threadIdx.x;
  for (int i = tid; i < B_ * DIMV; i += 256) {
    int g = i / DIMV, o = i % DIMV;
    float v = b1[o];
    for (int j = 0; j < 2 * DIMV; ++j) v += acc[g * 2 * DIMV + j] * W1[j * DIMV + o];
    hid[i] = v > 0.0f ? v : 0.0f;
  }
  __syncthreads();
  for (int i = tid; i < B_ * 6; i += 256) {
    int g = i / 6, o = i % 6;
    float v = b2[o];
    for (int j = 0; j < DIMV; ++j) v += hid[g * DIMV + j] * W2[j * 6 + o];
    out[i] = v;
  }
}

// ---------------------------------------------------------------------------
extern "C" void kernel_launch(void* const* d_in, const int* in_sizes, int n_in,
                              void* d_out, int out_size, void* d_ws, size_t ws_size,
                              hipStream_t stream) {
  const float* x = (const float*)d_in[0];
  const int* ei = (const int*)d_in[1];
  const int E = in_sizes[1] / 2;
  const int* src = ei;
  const int* dst = ei + E;

  const float* Wl[4] = {(const float*)d_in[3],  (const float*)d_in[7],
                        (const float*)d_in[11], (const float*)d_in[15]};
  const float* bl[4] = {(const float*)d_in[4],  (const float*)d_in[8],
                        (const float*)d_in[12], (const float*)d_in[16]};
  const float* Wr[4] = {(const float*)d_in[5],  (const float*)d_in[9],
                        (const float*)d_in[13], (const float*)d_in[17]};
  const float* pw[4] = {(const float*)d_in[6],  (const float*)d_in[10],
                        (const float*)d_in[14], (const float*)d_in[18]};
  const float* W1 = (const float*)d_in[19];
  const float* b1 = (const float*)d_in[20];
  const float* W2 = (const float*)d_in[21];
  const float* b2 = (const float*)d_in[22];
  float* out = (float*)d_out;

  // Workspace layout
  char* ws = (char*)d_ws;
  size_t off = 0;
  auto take = [&](size_t bytes) -> void* {
    void* p = (void*)(ws + off);
    off += (bytes + 255) & ~(size_t)255;
    return p;
  };
  float*    hA    = (float*)   take((size_t)BN_ * DIMV * 4);
  float*    hB    = (float*)   take((size_t)BN_ * DIMV * 4);
  float*    agg   = (float*)   take((size_t)BN_ * DIMV * 4);
  float*    deg   = (float*)   take((size_t)BN_ * 4);
  float*    sbuf  = (float*)   take((size_t)BN_ * 4);
  unsigned* key   = (unsigned*)take((size_t)BN_ * 4);
  int*      nmask = (int*)     take((size_t)BN_ * 4);
  float*    acc   = (float*)   take((size_t)B_ * 2 * DIMV * 4);
  unsigned* thr   = (unsigned*)take((size_t)B_ * 4);
  int*      keq   = (int*)     take((size_t)B_ * 4);
  (void)ws_size; (void)n_in; (void)out_size;

  const int KS[4] = {3277, 2622, 2098, 1679};

  init_state_kernel<<<(BN_ + 255) / 256, 256, 0, stream>>>(nmask, acc);

  const float* hin = x;
  float* bufs[2] = {hA, hB};
  for (int l = 0; l < 4; ++l) {
    float* hout = bufs[l & 1];

    zero_aggdeg_kernel<<<(BN_ * DIMV + 255) / 256, 256, 0, stream>>>(agg, deg);
    if (l == 0) {
      sage_aggregate_kernel<DIN0><<<(E + 255) / 256, 256, 0, stream>>>(src, dst, hin, nmask,
                                                                       agg, deg, E);
      sage_linear_kernel<DIN0><<<BN_ / 128, 256, 0, stream>>>(hin, agg, deg, Wl[l], bl[l],
                                                              Wr[l], hout);
    } else {
      sage_aggregate_kernel<DIMV><<<(E + 255) / 256, 256, 0, stream>>>(src, dst, hin, nmask,
                                                                       agg, deg, E);
      sage_linear_kernel<DIMV><<<BN_ / 128, 256, 0, stream>>>(hin, agg, deg, Wl[l], bl[l],
                                                              Wr[l], hout);
    }
    topk_score_kernel<<<BN_ / 256, 256, 0, stream>>>(hout, pw[l], nmask, sbuf, key);
    topk_radix_kernel<<<B_, 256, 0, stream>>>(key, KS[l], thr, keq);
    topk_apply_kernel<<<B_, 256, 0, stream>>>(key, thr, keq, sbuf, hout, nmask);
    readout_kernel<<<B_, 64, 0, stream>>>(hout, nmask, acc, 1.0f / (float)KS[l]);

    hin = hout;
  }

  mlp_kernel<<<1, 256, 0, stream>>>(acc, W1, b1, W2, b2, out);
}